// OmniRobotPhysics_29729763623475
// MI455X (gfx1250) — compile-verified
//
#include <hip/hip_runtime.h>
#include <cmath>

namespace {

constexpr float kDT      = 0.016f;
constexpr float kInvMass = 1.0f / 2.8f;
constexpr float kTwoPi   = 6.28318530717958647692f;
constexpr float kInv2Pi  = 0.15915494309189533577f;

// M = FORCE_MATRIX @ FORCE_MATRIX.T (constant, symmetric, M01 = M02 = 0)
constexpr float M00 =  2.6736481776f;
constexpr float M11 =  1.3263518224f;
constexpr float M12 = -0.0257017697f;
constexpr float M22 =  0.0324f;

struct Params {
  float dx, dy, invI;
  float g00, g11, g12, g22;           // gain*M (+grip on diagonal)
  float dv0, dv1, dv2, dc0, dc1, dc2; // softplus'd drag params
};

__device__ __forceinline__ float softplusf(float v) {
  // log1p(exp(v)), overflow-safe; matches jax.nn.softplus
  return fmaxf(v, 0.0f) + log1pf(__expf(-fabsf(v)));
}

__device__ __forceinline__ float sgnf(float v) {
  return (v > 0.0f) ? 1.0f : ((v < 0.0f) ? -1.0f : 0.0f);
}

__device__ __forceinline__ void dyn_step(float& x, float& y, float& th,
                                         float& vx, float& vy, float& om,
                                         float c0, float c1, float c2,
                                         const Params& p) {
  const float cs = __cosf(th);
  const float sn = __sinf(th);
  const float vxb =  vx * cs + vy * sn;
  const float vyb = -vx * sn + vy * cs;

  const float d0 = c0 - vxb, d1 = c1 - vyb, d2 = c2 - om;

  // F_geo = gain*(d @ M) + grip*d - drag_v*bv - drag_c*sign(bv)
  float Fx = p.g00 * d0              - p.dv0 * vxb - p.dc0 * sgnf(vxb);
  float Fy = p.g11 * d1 + p.g12 * d2 - p.dv1 * vyb - p.dc1 * sgnf(vyb);
  float Ft = p.g12 * d1 + p.g22 * d2 - p.dv2 * om  - p.dc2 * sgnf(om);

  const float tau = Ft - (p.dx * Fy - p.dy * Fx);
  const float ax  = Fx * kInvMass;
  const float ay  = Fy * kInvMass;
  const float aal = tau * p.invI;
  const float om2 = om * om;
  const float axb = ax - aal * p.dy - om2 * p.dx;
  const float ayb = ay + aal * p.dx - om2 * p.dy;
  const float axw = axb * cs - ayb * sn;
  const float ayw = axb * sn + ayb * cs;

  vx += axw * kDT;
  vy += ayw * kDT;
  om += aal * kDT;
  x  += vx * kDT;
  y  += vy * kDT;
  th += om * kDT;
  // atan2(sin(th), cos(th)) == wrap to [-pi, pi)
  th -= kTwoPi * rintf(th * kInv2Pi);
}

__device__ __forceinline__ void store_state(float* op, float x, float y, float th,
                                            float vx, float vy, float om) {
  // rows are only 8B-aligned (24B stride) -> float2 (global_store_b64) x3
  reinterpret_cast<float2*>(op)[0] = make_float2(x,  y);
  reinterpret_cast<float2*>(op)[1] = make_float2(th, vx);
  reinterpret_cast<float2*>(op)[2] = make_float2(vy, om);
}

// 4 scan steps (one 48B command chunk) for NR interleaved robots.
template <int NR>
__device__ __forceinline__ void do_chunk(float (&x)[NR], float (&y)[NR], float (&th)[NR],
                                         float (&vx)[NR], float (&vy)[NR], float (&om)[NR],
                                         float4 (&C0)[NR], float4 (&C1)[NR], float4 (&C2)[NR],
                                         float* (&op)[NR], const Params& p) {
#pragma unroll
  for (int k = 0; k < NR; ++k) {
    dyn_step(x[k], y[k], th[k], vx[k], vy[k], om[k], C0[k].x, C0[k].y, C0[k].z, p);
    store_state(op[k], x[k], y[k], th[k], vx[k], vy[k], om[k]); op[k] += 6;
  }
#pragma unroll
  for (int k = 0; k < NR; ++k) {
    dyn_step(x[k], y[k], th[k], vx[k], vy[k], om[k], C0[k].w, C1[k].x, C1[k].y, p);
    store_state(op[k], x[k], y[k], th[k], vx[k], vy[k], om[k]); op[k] += 6;
  }
#pragma unroll
  for (int k = 0; k < NR; ++k) {
    dyn_step(x[k], y[k], th[k], vx[k], vy[k], om[k], C1[k].z, C1[k].w, C2[k].x, p);
    store_state(op[k], x[k], y[k], th[k], vx[k], vy[k], om[k]); op[k] += 6;
  }
#pragma unroll
  for (int k = 0; k < NR; ++k) {
    dyn_step(x[k], y[k], th[k], vx[k], vy[k], om[k], C2[k].y, C2[k].z, C2[k].w, p);
    store_state(op[k], x[k], y[k], th[k], vx[k], vy[k], om[k]); op[k] += 6;
  }
}

// NR independent scan chains per thread, interleaved so the SIMD fills one
// chain's dependency bubbles with the other. Command stream is ping-pong
// double-buffered (unroll-by-2) so refills load straight into the live
// buffer registers -- no reg-reg copies at the loop backedge.
template <int NR>
__device__ __forceinline__ void run_scan(int tid, int half,
                                         const float* __restrict__ init,
                                         const float* __restrict__ cmds,
                                         float* __restrict__ out,
                                         const Params& p, int T) {
  float x[NR], y[NR], th[NR], vx[NR], vy[NR], om[NR];
  const float*  cp[NR];
  const float4* cp4[NR];
  float* op[NR];

#pragma unroll
  for (int k = 0; k < NR; ++k) {
    const int b = tid + k * half;
    const float2* ip = reinterpret_cast<const float2*>(init + (size_t)b * 6);
    const float2 s0 = ip[0], s1 = ip[1], s2 = ip[2];
    x[k] = s0.x; y[k] = s0.y; th[k] = s1.x;
    vx[k] = s1.y; vy[k] = s2.x; om[k] = s2.y;

    op[k] = out + (size_t)b * (size_t)(T + 1) * 6;
    reinterpret_cast<float2*>(op[k])[0] = s0;
    reinterpret_cast<float2*>(op[k])[1] = s1;
    reinterpret_cast<float2*>(op[k])[2] = s2;
    op[k] += 6;

    cp[k]  = cmds + (size_t)b * (size_t)T * 3;
    cp4[k] = reinterpret_cast<const float4*>(cp[k]);
  }

  const int nch = T >> 2;            // 4 steps (48B = 3 x float4) per chunk
  float4 A0[NR], A1[NR], A2[NR];     // buffer set A: even chunks
  float4 B0[NR], B1[NR], B2[NR];     // buffer set B: odd chunks
#pragma unroll
  for (int k = 0; k < NR; ++k) {
    A0[k] = {}; A1[k] = {}; A2[k] = {};
    B0[k] = {}; B1[k] = {}; B2[k] = {};
  }
  if (nch > 0) {
#pragma unroll
    for (int k = 0; k < NR; ++k) { A0[k] = cp4[k][0]; A1[k] = cp4[k][1]; A2[k] = cp4[k][2]; }
  }
  if (nch > 1) {
#pragma unroll
    for (int k = 0; k < NR; ++k) { B0[k] = cp4[k][3]; B1[k] = cp4[k][4]; B2[k] = cp4[k][5]; }
  }

  int i = 0;
  for (; i + 1 < nch; i += 2) {
#pragma unroll
    for (int k = 0; k < NR; ++k)
      __builtin_prefetch(cp[k] + 12 * (i + 8), 0, 3);  // gfx1250 global_prefetch_b8

    // chunk i from A, then refill A with chunk i+2 (in flight while B computes)
    do_chunk<NR>(x, y, th, vx, vy, om, A0, A1, A2, op, p);
    if (i + 2 < nch) {
#pragma unroll
      for (int k = 0; k < NR; ++k) {
        const float4* q = cp4[k] + 3 * (i + 2);
        A0[k] = q[0]; A1[k] = q[1]; A2[k] = q[2];
      }
    }

    // chunk i+1 from B, then refill B with chunk i+3
    do_chunk<NR>(x, y, th, vx, vy, om, B0, B1, B2, op, p);
    if (i + 3 < nch) {
#pragma unroll
      for (int k = 0; k < NR; ++k) {
        const float4* q = cp4[k] + 3 * (i + 3);
        B0[k] = q[0]; B1[k] = q[1]; B2[k] = q[2];
      }
    }
  }

  // odd leftover chunk sits in A
  if (i < nch) {
    do_chunk<NR>(x, y, th, vx, vy, om, A0, A1, A2, op, p);
    ++i;
  }

  // tail (T % 4 != 0); T=1024 in practice so this is empty
  for (int t = nch * 4; t < T; ++t) {
#pragma unroll
    for (int k = 0; k < NR; ++k) {
      dyn_step(x[k], y[k], th[k], vx[k], vy[k], om[k],
               cp[k][3 * t], cp[k][3 * t + 1], cp[k][3 * t + 2], p);
      store_state(op[k], x[k], y[k], th[k], vx[k], vy[k], om[k]); op[k] += 6;
    }
  }
}

__global__ void __launch_bounds__(64, 1)
omni_scan_kernel(const float* __restrict__ init,
                 const float* __restrict__ cmds,
                 const float* __restrict__ com,
                 const float* __restrict__ inertia_p,
                 const float* __restrict__ gain_p,
                 const float* __restrict__ grip_p,
                 const float* __restrict__ dragv,
                 const float* __restrict__ dragc,
                 float* __restrict__ out,
                 int B, int T) {
  const int tid  = blockIdx.x * blockDim.x + threadIdx.x;
  const int half = (B + 1) >> 1;
  if (tid >= half) return;

  // Wave-uniform parameter prep: scalarized onto the gfx1250 scalar-float
  // ALU (s_mul_f32 / v_s_exp_f32 path); results live in SGPRs.
  const float gain = softplusf(gain_p[0]);
  const float grip = softplusf(grip_p[0]);
  Params p;
  p.dx   = com[0];
  p.dy   = com[1];
  p.invI = 1.0f / (softplusf(inertia_p[0]) + 1e-4f);
  p.g00  = gain * M00 + grip;
  p.g11  = gain * M11 + grip;
  p.g12  = gain * M12;
  p.g22  = gain * M22 + grip;
  p.dv0 = softplusf(dragv[0]); p.dv1 = softplusf(dragv[1]); p.dv2 = softplusf(dragv[2]);
  p.dc0 = softplusf(dragc[0]); p.dc1 = softplusf(dragc[1]); p.dc2 = softplusf(dragc[2]);

  if (tid + half < B) {
    run_scan<2>(tid, half, init, cmds, out, p, T);   // two interleaved chains
  } else {
    run_scan<1>(tid, half, init, cmds, out, p, T);   // odd-B straggler
  }
}

} // namespace

extern "C" void kernel_launch(void* const* d_in, const int* in_sizes, int n_in,
                              void* d_out, int out_size, void* d_ws, size_t ws_size,
                              hipStream_t stream) {
  const float* init = (const float*)d_in[0];  // (B, 6)
  const float* cmds = (const float*)d_in[1];  // (B, T, 3)
  const float* com  = (const float*)d_in[2];  // (2,)
  const float* inrt = (const float*)d_in[3];  // scalar
  const float* gain = (const float*)d_in[4];  // scalar
  const float* grip = (const float*)d_in[5];  // scalar
  const float* dv   = (const float*)d_in[6];  // (3,)
  const float* dc   = (const float*)d_in[7];  // (3,)
  float* out = (float*)d_out;                 // (B, T+1, 6)

  const int B = in_sizes[0] / 6;
  const int T = (B > 0) ? in_sizes[1] / (3 * B) : 0;
  const int half = (B + 1) >> 1;              // threads; 2 robots per thread

  dim3 block(64);
  dim3 grid((half + (int)block.x - 1) / (int)block.x);
  hipLaunchKernelGGL(omni_scan_kernel, grid, block, 0, stream,
                     init, cmds, com, inrt, gain, grip, dv, dc, out, B, T);
}